// SRGNN_20779051778106
// MI455X (gfx1250) — compile-verified
//
#include <hip/hip_runtime.h>
#include <hip/hip_bf16.h>
#include <math.h>

#define DEMB 64

typedef __attribute__((ext_vector_type(2))) float v2f;
typedef __attribute__((ext_vector_type(8))) float v8f;

// f32 WMMA: D(16x16) = A(16x4) * B(4x16) + C, exact fp32 math (matches reference dtype)
__device__ __forceinline__ v8f wmma4(v2f a, v2f b, v8f c) {
  return __builtin_amdgcn_wmma_f32_16x16x4_f32(false, a, false, b, (short)0, c, false, false);
}

__device__ __forceinline__ float sigm(float x) { return 1.0f / (1.0f + expf(-x)); }

// ---------- vectorized streaming kernels ----------

__global__ void k_zero4(float4* __restrict__ p, long n4) {
  long i = (long)blockIdx.x * blockDim.x + threadIdx.x;
  if (i < n4) p[i] = make_float4(0.f, 0.f, 0.f, 0.f);
}

__global__ void k_init_i32(int* __restrict__ p, int n, int v) {
  int i = blockIdx.x * blockDim.x + threadIdx.x;
  if (i < n) p[i] = v;
}

// h[n,:] = item_emb[node_items[n]-1,:]  (float4 lanes: 16 per row)
__global__ void k_gather4(const float4* __restrict__ item_emb, const int* __restrict__ node_items,
                          float4* __restrict__ h, long n16) {
  long i = (long)blockIdx.x * blockDim.x + threadIdx.x;
  if (i >= n16) return;
  int n = (int)(i >> 4), q = (int)(i & 15);
  h[i] = item_emb[(long)(node_items[n] - 1) * 16 + q];
}

__global__ void k_relu4(float4* __restrict__ h, long n4) {
  long i = (long)blockIdx.x * blockDim.x + threadIdx.x;
  if (i >= n4) return;
  float4 v = h[i];
  v.x = fmaxf(v.x, 0.f); v.y = fmaxf(v.y, 0.f);
  v.z = fmaxf(v.z, 0.f); v.w = fmaxf(v.w, 0.f);
  h[i] = v;
}

// agg[dst,:] += m[src,:]  (chain graph: one in-edge per dst -> atomics are order-free)
__global__ void k_scatter4(const float4* __restrict__ m, const int* __restrict__ esrc,
                           const int* __restrict__ edst, float* __restrict__ agg, long e16) {
  long i = (long)blockIdx.x * blockDim.x + threadIdx.x;
  if (i >= e16) return;
  int e = (int)(i >> 4), q = (int)(i & 15);
  float4 v = m[(long)esrc[e] * 16 + q];
  float* dst = agg + (long)edst[e] * DEMB + q * 4;
  atomicAdd(dst + 0, v.x); atomicAdd(dst + 1, v.y);
  atomicAdd(dst + 2, v.z); atomicAdd(dst + 3, v.w);
}

// segment_max(arange, batch): integer atomics (order-independent -> deterministic)
__global__ void k_lastidx(const int* __restrict__ batch, int* __restrict__ last_idx, int N) {
  int n = blockIdx.x * blockDim.x + threadIdx.x;
  if (n < N) atomicMax(&last_idx[batch[n]], n);
}

// s_g[s,d4] = sum_l alpha[s*L+l] * h[s*L+l, d4]  (serial over L -> deterministic)
__global__ void k_sg4(const float4* __restrict__ h, const float* __restrict__ alpha,
                      float4* __restrict__ s_g, long s16, int L) {
  long i = (long)blockIdx.x * blockDim.x + threadIdx.x;
  if (i >= s16) return;
  int s = (int)(i >> 4), q = (int)(i & 15);
  long nb = (long)s * L;
  float4 acc = make_float4(0.f, 0.f, 0.f, 0.f);
  for (int l = 0; l < L; ++l) {
    float a = alpha[nb + l];
    float4 v = h[(nb + l) * 16 + q];
    acc.x += a * v.x; acc.y += a * v.y; acc.z += a * v.z; acc.w += a * v.w;
  }
  s_g[i] = acc;
}

// ---------- WMMA GEMM kernels ----------

// C[N,64] = A[N,64] @ Wg[64,64]. Persistent waves: B fragments register-resident
// (16 k-steps x 4 col-tiles x v2f = 128 VGPRs), grid-stride over 16-row tiles.
__global__ __launch_bounds__(256)
void k_gemm_m(const float* __restrict__ A, const float* __restrict__ Wg,
              float* __restrict__ C, int ntiles) {
  int lane = threadIdx.x & 31;
  int half = lane >> 4, l16 = lane & 15;

  v2f bfrag[16][4];
#pragma unroll
  for (int ki = 0; ki < 16; ++ki) {
    int kk = ki * 4 + 2 * half;
#pragma unroll
    for (int ct = 0; ct < 4; ++ct) {
      const float* bp = Wg + kk * DEMB + ct * 16 + l16;   // B[k,j] = Wg[k*64+j]
      bfrag[ki][ct].x = bp[0];
      bfrag[ki][ct].y = bp[DEMB];
    }
  }

  int wave0  = (int)((blockIdx.x * blockDim.x + threadIdx.x) >> 5);
  int nwaves = (int)((gridDim.x * blockDim.x) >> 5);
  for (int tile = wave0; tile < ntiles; tile += nwaves) {   // wave-uniform loop: EXEC all-ones
    int base = tile << 4;
    v8f acc[4] = {};
    for (int ki = 0; ki < 16; ++ki) {
      const float* ap = A + (long)(base + l16) * DEMB + ki * 4 + 2 * half;
      v2f a; a.x = ap[0]; a.y = ap[1];
#pragma unroll
      for (int ct = 0; ct < 4; ++ct) acc[ct] = wmma4(a, bfrag[ki][ct], acc[ct]);
    }
#pragma unroll
    for (int ct = 0; ct < 4; ++ct) {
      int col = ct * 16 + l16;
#pragma unroll
      for (int r = 0; r < 8; ++r)
        C[(long)(base + half * 8 + r) * DEMB + col] = acc[ct][r];
    }
  }
}

// Fused GRUCell, fully register-resident: gi/gh accumulators (12+12 tiles, 192 VGPRs),
// gate math done in-register exploiting tile-column alignment (ct, ct+4, ct+8).
// No LDS, no barriers. Block touches only its own 16 rows -> in-place h update safe.
__global__ __launch_bounds__(32)
void k_gru(const float* __restrict__ agg, float* h,
           const float* __restrict__ W_ih, const float* __restrict__ b_ih,
           const float* __restrict__ W_hh, const float* __restrict__ b_hh) {
  int base = blockIdx.x << 4;
  int lane = threadIdx.x & 31;
  int half = lane >> 4, l16 = lane & 15;

  v8f gi[12], gh[12];
  v8f zero = {};
#pragma unroll
  for (int ct = 0; ct < 12; ++ct) { gi[ct] = zero; gh[ct] = zero; }

  for (int ki = 0; ki < 16; ++ki) {
    int kk = ki * 4 + 2 * half;
    v2f aa, ah;
    { const float* p = agg + (long)(base + l16) * DEMB + kk; aa.x = p[0]; aa.y = p[1]; }
    { const float* p = h   + (long)(base + l16) * DEMB + kk; ah.x = p[0]; ah.y = p[1]; }
#pragma unroll
    for (int ct = 0; ct < 12; ++ct) {
      int col = ct * 16 + l16;
      v2f bi, bh;                                   // B[k,j] = W[j*64+k]: adjacent pair -> b64
      { const float* p = W_ih + (long)col * DEMB + kk; bi.x = p[0]; bi.y = p[1]; }
      { const float* p = W_hh + (long)col * DEMB + kk; bh.x = p[0]; bh.y = p[1]; }
      gi[ct] = wmma4(aa, bi, gi[ct]);
      gh[ct] = wmma4(ah, bh, gh[ct]);
    }
  }

#pragma unroll
  for (int ct = 0; ct < 4; ++ct) {
    int col = ct * 16 + l16;
    float bir = b_ih[col], biz = b_ih[64 + col], bin = b_ih[128 + col];
    float bhr = b_hh[col], bhz = b_hh[64 + col], bhn = b_hh[128 + col];
#pragma unroll
    for (int r = 0; r < 8; ++r) {
      float i_r = gi[ct][r] + bir, i_z = gi[ct + 4][r] + biz, i_n = gi[ct + 8][r] + bin;
      float h_r = gh[ct][r] + bhr, h_z = gh[ct + 4][r] + bhz, h_n = gh[ct + 8][r] + bhn;
      float rg = sigm(i_r + h_r);
      float zg = sigm(i_z + h_z);
      float ng = tanhf(i_n + rg * h_n);
      long idx = (long)(base + half * 8 + r) * DEMB + col;
      float hp = h[idx];
      h[idx] = (1.0f - zg) * ng + zg * hp;
    }
  }
}

// Fused attention gate: pre = v_n[batch]@W1^T + h@W2^T + b1+b2; alpha = sigmoid(pre)@q_w^T + q_b.
// Two WMMA chains share one accumulator; alpha reduced with shfl_xor butterflies (no LDS).
__global__ __launch_bounds__(32)
void k_gate(const float* __restrict__ h, const int* __restrict__ batch,
            const int* __restrict__ last_idx,
            const float* __restrict__ W1, const float* __restrict__ b1,
            const float* __restrict__ W2, const float* __restrict__ b2,
            const float* __restrict__ q_w, const float* __restrict__ q_b,
            float* __restrict__ alpha) {
  int base = blockIdx.x << 4;
  int lane = threadIdx.x & 31;
  int half = lane >> 4, l16 = lane & 15;
  int arow = base + l16;
  long vrow = last_idx[batch[arow]];

  v8f acc[4] = {};
  for (int ki = 0; ki < 16; ++ki) {
    int kk = ki * 4 + 2 * half;
    v2f a1, a2;
    { const float* p = h + vrow * DEMB + kk;         a1.x = p[0]; a1.y = p[1]; }
    { const float* p = h + (long)arow * DEMB + kk;   a2.x = p[0]; a2.y = p[1]; }
#pragma unroll
    for (int ct = 0; ct < 4; ++ct) {
      int col = ct * 16 + l16;
      v2f bw1, bw2;
      { const float* p = W1 + (long)col * DEMB + kk; bw1.x = p[0]; bw1.y = p[1]; }
      { const float* p = W2 + (long)col * DEMB + kk; bw2.x = p[0]; bw2.y = p[1]; }
      acc[ct] = wmma4(a1, bw1, acc[ct]);
      acc[ct] = wmma4(a2, bw2, acc[ct]);
    }
  }

  float pal[8];
#pragma unroll
  for (int r = 0; r < 8; ++r) pal[r] = 0.0f;
#pragma unroll
  for (int ct = 0; ct < 4; ++ct) {
    int col = ct * 16 + l16;
    float bias = b1[col] + b2[col];
    float qw = q_w[col];
#pragma unroll
    for (int r = 0; r < 8; ++r) pal[r] += sigm(acc[ct][r] + bias) * qw;
  }
  // reduce across the 16 lanes of each half-group (xor < 16 stays in-group)
#pragma unroll
  for (int m = 1; m < 16; m <<= 1)
#pragma unroll
    for (int r = 0; r < 8; ++r) pal[r] += __shfl_xor(pal[r], m, 32);
  if (l16 == 0) {
    float qb = q_b[0];
#pragma unroll
    for (int r = 0; r < 8; ++r) alpha[base + half * 8 + r] = pal[r] + qb;
  }
}

// s_h = concat(v_n, s_g) @ W3^T + b3  (K=128; A-fragment loader handles the concat)
__global__ __launch_bounds__(256)
void k_sh(const float* __restrict__ h, const int* __restrict__ last_idx,
          const float* __restrict__ s_g, const float* __restrict__ W3,
          const float* __restrict__ b3, float* __restrict__ s_h, int stiles) {
  int wave = (int)((blockIdx.x * blockDim.x + threadIdx.x) >> 5);
  if (wave >= stiles) return;                  // wave-granular guard
  int lane = threadIdx.x & 31;
  int half = lane >> 4, l16 = lane & 15;
  int base = wave << 4;
  int srow = base + l16;
  long vrow = last_idx[srow];
  v8f acc[4] = {};
  for (int k0 = 0; k0 < 2 * DEMB; k0 += 4) {
    int kk = k0 + 2 * half;
    v2f a;
    if (kk < DEMB) { const float* p = h + vrow * DEMB + kk;                  a.x = p[0]; a.y = p[1]; }
    else           { const float* p = s_g + (long)srow * DEMB + (kk - DEMB); a.x = p[0]; a.y = p[1]; }
#pragma unroll
    for (int ct = 0; ct < 4; ++ct) {
      int col = ct * 16 + l16;
      const float* q = W3 + (long)col * (2 * DEMB) + kk;   // B[k,j] = W3[j*128+k]
      v2f b; b.x = q[0]; b.y = q[1];
      acc[ct] = wmma4(a, b, acc[ct]);
    }
  }
#pragma unroll
  for (int ct = 0; ct < 4; ++ct) {
    int col = ct * 16 + l16;
    float bias = b3[col];
#pragma unroll
    for (int r = 0; r < 8; ++r)
      s_h[(long)(base + half * 8 + r) * DEMB + col] = acc[ct][r] + bias;
  }
}

// ---------- loss + deterministic reduction ----------

__global__ __launch_bounds__(256)
void k_loss(const float* __restrict__ s_h, const float* __restrict__ item_emb,
            const int* __restrict__ pos, const int* __restrict__ neg,
            float* __restrict__ partials, int S) {
  __shared__ float red[4 * 256];
  int s = blockIdx.x * 256 + threadIdx.x;
  float lo = 0.0f, sh2 = 0.0f, pe2 = 0.0f, ne2 = 0.0f;
  if (s < S) {
    const float4* sh = (const float4*)(s_h + (long)s * DEMB);
    const float4* pe = (const float4*)(item_emb + (long)pos[s] * DEMB);
    const float4* ne = (const float4*)(item_emb + (long)neg[s] * DEMB);
    float pp = 0.0f, np = 0.0f;
    for (int d = 0; d < 16; ++d) {
      float4 a = sh[d], b = pe[d], c = ne[d];
      pp  += a.x * b.x + a.y * b.y + a.z * b.z + a.w * b.w;
      np  += a.x * c.x + a.y * c.y + a.z * c.z + a.w * c.w;
      sh2 += a.x * a.x + a.y * a.y + a.z * a.z + a.w * a.w;
      pe2 += b.x * b.x + b.y * b.y + b.z * b.z + b.w * b.w;
      ne2 += c.x * c.x + c.y * c.y + c.z * c.z + c.w * c.w;
    }
    float t = -(pp - np);                                   // -log_sigmoid(x) = softplus(-x)
    lo = fmaxf(t, 0.0f) + log1pf(expf(-fabsf(t)));
  }
  red[threadIdx.x] = lo; red[256 + threadIdx.x] = sh2;
  red[512 + threadIdx.x] = pe2; red[768 + threadIdx.x] = ne2;
  __syncthreads();
  for (int st = 128; st > 0; st >>= 1) {                    // fixed tree -> deterministic
    if ((int)threadIdx.x < st) {
      red[threadIdx.x]       += red[threadIdx.x + st];
      red[256 + threadIdx.x] += red[256 + threadIdx.x + st];
      red[512 + threadIdx.x] += red[512 + threadIdx.x + st];
      red[768 + threadIdx.x] += red[768 + threadIdx.x + st];
    }
    __syncthreads();
  }
  if (threadIdx.x == 0) {
    partials[blockIdx.x * 4 + 0] = red[0];
    partials[blockIdx.x * 4 + 1] = red[256];
    partials[blockIdx.x * 4 + 2] = red[512];
    partials[blockIdx.x * 4 + 3] = red[768];
  }
}

__global__ void k_final(const float* __restrict__ partials, int nb, float* __restrict__ out) {
  if (blockIdx.x == 0 && threadIdx.x == 0) {
    float lo = 0.0f, sh2 = 0.0f, pe2 = 0.0f, ne2 = 0.0f;
    for (int i = 0; i < nb; ++i) {
      lo  += partials[i * 4 + 0];
      sh2 += partials[i * 4 + 1];
      pe2 += partials[i * 4 + 2];
      ne2 += partials[i * 4 + 3];
    }
    float reg = 1e-5f * (sh2 + pe2 + ne2);
    out[0] = lo + reg; out[1] = lo; out[2] = reg; out[3] = reg; out[4] = reg;
  }
}

// ---------- host orchestration ----------

static inline int cdiv(long a, long b) { return (int)((a + b - 1) / b); }

extern "C" void kernel_launch(void* const* d_in, const int* in_sizes, int n_in,
                              void* d_out, int out_size, void* d_ws, size_t ws_size,
                              hipStream_t stream) {
  const float* item_emb = (const float*)d_in[0];
  const float* ggc_w    = (const float*)d_in[1];   // [2,64,64]
  const float* W_ih     = (const float*)d_in[2];
  const float* b_ih     = (const float*)d_in[3];
  const float* W_hh     = (const float*)d_in[4];
  const float* b_hh     = (const float*)d_in[5];
  const float* W1       = (const float*)d_in[6];
  const float* b1       = (const float*)d_in[7];
  const float* W2       = (const float*)d_in[8];
  const float* b2       = (const float*)d_in[9];
  const float* q_w      = (const float*)d_in[10];
  const float* q_b      = (const float*)d_in[11];
  const float* W3       = (const float*)d_in[12];
  const float* b3       = (const float*)d_in[13];
  const int* node_items = (const int*)d_in[14];
  const int* edge_index = (const int*)d_in[15];
  const int* batch      = (const int*)d_in[16];
  const int* pos        = (const int*)d_in[17];
  const int* rneg       = (const int*)d_in[18];

  const int N = in_sizes[14];
  const int E = in_sizes[15] / 2;
  const int S = in_sizes[17];
  const int L = N / S;
  const int* esrc = edge_index;
  const int* edst = edge_index + E;

  // workspace carve-out (256B aligned)
  size_t off = 0;
  auto alloc = [&](size_t bytes) -> void* {
    off = (off + 255) & ~(size_t)255;
    void* p = (char*)d_ws + off;
    off += bytes;
    return p;
  };
  float* h        = (float*)alloc((size_t)N * DEMB * 4);
  float* m        = (float*)alloc((size_t)N * DEMB * 4);
  float* agg      = (float*)alloc((size_t)N * DEMB * 4);
  float* alpha    = (float*)alloc((size_t)N * 4);
  float* s_g      = (float*)alloc((size_t)S * DEMB * 4);
  float* s_hbuf   = (float*)alloc((size_t)S * DEMB * 4);
  int lossBlocks  = cdiv(S, 256);
  float* partials = (float*)alloc((size_t)lossBlocks * 4 * 4);
  int* last_idx   = (int*)alloc((size_t)S * 4);
  (void)ws_size; (void)n_in; (void)out_size;

  const int ntiles = N / 16;    // N = S*L, multiple of 16 for this workload
  const int stiles = S / 16;
  const long N16 = (long)N * 16;   // float4 count per [N,64]
  const long S16 = (long)S * 16;
  const long E16 = (long)E * 16;

  // 1) embedding gather
  k_gather4<<<cdiv(N16, 256), 256, 0, stream>>>((const float4*)item_emb, node_items, (float4*)h, N16);

  // 2) two GatedGraphConv layers (shared GRUCell)
  const int gemmBlocks = 512;   // 4096 persistent waves, ~5 tiles each
  for (int layer = 0; layer < 2; ++layer) {
    const float* Wg = ggc_w + (size_t)layer * DEMB * DEMB;
    k_gemm_m<<<gemmBlocks, 256, 0, stream>>>(h, Wg, m, ntiles);
    k_zero4<<<cdiv(N16, 256), 256, 0, stream>>>((float4*)agg, N16);
    k_scatter4<<<cdiv(E16, 256), 256, 0, stream>>>((const float4*)m, esrc, edst, agg, E16);
    k_gru<<<ntiles, 32, 0, stream>>>(agg, h, W_ih, b_ih, W_hh, b_hh);
  }
  k_relu4<<<cdiv(N16, 256), 256, 0, stream>>>((float4*)h, N16);

  // 3) session attention
  k_init_i32<<<cdiv(S, 256), 256, 0, stream>>>(last_idx, S, 0);
  k_lastidx<<<cdiv(N, 256), 256, 0, stream>>>(batch, last_idx, N);
  k_gate<<<ntiles, 32, 0, stream>>>(h, batch, last_idx, W1, b1, W2, b2, q_w, q_b, alpha);
  k_sg4<<<cdiv(S16, 256), 256, 0, stream>>>((const float4*)h, alpha, (float4*)s_g, S16, L);
  k_sh<<<cdiv((long)stiles * 32, 256), 256, 0, stream>>>(h, last_idx, s_g, W3, b3, s_hbuf, stiles);

  // 4) BPR loss + regularizer
  k_loss<<<lossBlocks, 256, 0, stream>>>(s_hbuf, item_emb, pos, rneg, partials, S);
  k_final<<<1, 32, 0, stream>>>(partials, lossBlocks, (float*)d_out);
}